// Net_38482906972413
// MI455X (gfx1250) — compile-verified
//
#include <hip/hip_runtime.h>
#include <stdint.h>

typedef __attribute__((ext_vector_type(2))) float v2f;
typedef __attribute__((ext_vector_type(4))) float v4f;
typedef __attribute__((ext_vector_type(8))) float v8f;
typedef __attribute__((ext_vector_type(4))) int   v4i;

// Problem sizes (fixed by the reference)
constexpr int Bn   = 32;
constexpr int Nn   = 128;
constexpr int En   = 8192;
constexpr int IMGc = 2048;
constexpr int LOCc = 5;
constexpr int Hc   = 128;

// ---------------------------------------------------------------------------
// Async global->LDS copy (GLOBAL_LOAD_ASYNC_TO_LDS_B128, ASYNCcnt-tracked).
// ROCm 7.2 declares the builtin with v4i-pointee pointer params; cast through
// uintptr_t to AS(1)/AS(3)-qualified v4i pointers. Falls back to a
// synchronous v4f copy if the builtin is absent.
// ---------------------------------------------------------------------------
#if __has_builtin(__builtin_amdgcn_global_load_async_to_lds_b128)
#define HAS_ASYNC_LDS 1
#else
#define HAS_ASYNC_LDS 0
#endif

__device__ __forceinline__ void copy_b128_to_lds(const float* gsrc, float* ldst) {
#if HAS_ASYNC_LDS
    __attribute__((address_space(1))) v4i* g =
        (__attribute__((address_space(1))) v4i*)(uintptr_t)gsrc;
    // low 32 bits of a flat shared-aperture address are the LDS byte offset
    __attribute__((address_space(3))) v4i* l =
        (__attribute__((address_space(3))) v4i*)(uintptr_t)(unsigned)(uintptr_t)ldst;
    __builtin_amdgcn_global_load_async_to_lds_b128(g, l, 0, 0);
#else
    *(v4f*)ldst = *(const v4f*)gsrc;
#endif
}

__device__ __forceinline__ void wait_async_lds() {
#if HAS_ASYNC_LDS
#if __has_builtin(__builtin_amdgcn_s_wait_asynccnt)
    __builtin_amdgcn_s_wait_asynccnt(0);
#else
    asm volatile("s_wait_asynccnt 0x0" ::: "memory");
#endif
#endif
}

// ---------------------------------------------------------------------------
// Kernel 1: img = img_feat @ W_img + b_img   (stored to workspace)
//           node_feat = img + (img_loc @ W_loc + b_loc)  (stored to d_out)
// One block = 16 rows x 128 cols; 8 waves, one 16x16 WMMA tile per wave.
// A double-buffered through LDS via async copies: chunk c+1 fetch overlaps
// chunk c WMMA; one barrier per chunk. Row stride 68 dwords (== 4 mod 64)
// keeps the ds_load_2addr_b64 A-fragment reads bank-conflict-free.
// ---------------------------------------------------------------------------
__launch_bounds__(256, 2)
__global__ void node_proj_kernel(const float* __restrict__ img_feat,
                                 const float* __restrict__ img_loc,
                                 const float* __restrict__ W_img,
                                 const float* __restrict__ b_img,
                                 const float* __restrict__ W_loc,
                                 const float* __restrict__ b_loc,
                                 float* __restrict__ img_ws,
                                 float* __restrict__ node_out) {
    constexpr int KC     = 64;          // K chunk per stage
    constexpr int LD     = 68;          // LDS row stride (dwords)
    constexpr int CHUNKS = IMGc / KC;   // 32
    __shared__ float As[2][16 * LD];    // double buffer, 8.5 KB

    const int tid  = threadIdx.x;
    const int wave = tid >> 5;
    const int lane = tid & 31;
    const int ln   = lane & 15;   // A row within tile / B,C column
    const int hi   = lane >> 4;   // half-wave select (K phase / M offset)
    const int m0   = blockIdx.x * 16;   // row block in [0, B*N)
    const int n0   = wave * 16;         // column block in [0, H)

    // staging coordinates: 16 threads per row, one b128 each
    const int lr = tid >> 4;          // 0..15
    const int lc = (tid & 15) * 4;    // 0..60
    const float* gA = img_feat + (size_t)(m0 + lr) * IMGc + lc;

    // prologue: stage chunk 0 into buffer 0
    copy_b128_to_lds(gA, &As[0][lr * LD + lc]);

    v8f acc = {};

    for (int c = 0; c < CHUNKS; ++c) {
        const int cur = c & 1;
        wait_async_lds();          // chunk c has landed in LDS (this wave)
        __syncthreads();           // ... and everyone else's portion too;
                                   // also: all waves done reading buf[cur^1]
        if (c + 1 < CHUNKS)        // overlap: fetch chunk c+1 during compute
            copy_b128_to_lds(gA + (c + 1) * KC, &As[cur ^ 1][lr * LD + lc]);

        const int k0 = c * KC;
        #pragma unroll
        for (int kk = 0; kk < KC; kk += 4) {
            // A fragment: lane ln holds row m0+ln, K = kk+2*hi, kk+1+2*hi
            v2f a = *(const v2f*)(&As[cur][ln * LD + kk + 2 * hi]);
            // B fragment: lane ln holds col n0+ln, same K pair
            v2f bf;
            bf.x = W_img[(size_t)(k0 + kk     + 2 * hi) * Hc + n0 + ln];
            bf.y = W_img[(size_t)(k0 + kk + 1 + 2 * hi) * Hc + n0 + ln];
            acc = __builtin_amdgcn_wmma_f32_16x16x4_f32(
                      false, a, false, bf, (short)0, acc, false, false);
        }
    }

    // Epilogue: bias + tiny loc projection (K=5) in VALU, then store.
    const int   col = n0 + ln;
    const float bi  = b_img[col];
    const float bl  = b_loc[col];
    float wl[LOCc];
    #pragma unroll
    for (int i = 0; i < LOCc; ++i) wl[i] = W_loc[i * Hc + col];

    #pragma unroll
    for (int j = 0; j < 8; ++j) {
        const int row   = m0 + j + 8 * hi;        // C layout: M = j + 8*hi
        const float img_v = acc[j] + bi;
        float loc_v = bl;
        #pragma unroll
        for (int i = 0; i < LOCc; ++i)
            loc_v += img_loc[(size_t)row * LOCc + i] * wl[i];
        img_ws[(size_t)row * Hc + col]   = img_v;          // for edge kernel
        node_out[(size_t)row * Hc + col] = img_v + loc_v;  // node_feat
    }
}

// ---------------------------------------------------------------------------
// Kernel 2: edge_feat[b,e,:] = img[b,src[e],:] @ W_rel[0:H]
//                            + img[b,dst[e],:] @ W_rel[H:2H] + b_rel
// One block = 16 edges x 128 cols for one batch b. The gathered src|dst rows
// are staged once (async) into a 16x256 LDS tile (concat matches W_rel's 256
// rows), shared by all 8 waves => 8x fewer gathered loads. Row stride 260
// dwords (== 4 mod 64) keeps A-fragment b64 reads bank-conflict-free.
// ---------------------------------------------------------------------------
__launch_bounds__(256, 2)
__global__ void edge_kernel(const float* __restrict__ img_ws,
                            const int*   __restrict__ edge_src,
                            const int*   __restrict__ edge_dst,
                            const float* __restrict__ W_rel,
                            const float* __restrict__ b_rel,
                            float* __restrict__ edge_out) {
    constexpr int LD = 260;              // 256 + 4 pad dwords
    __shared__ float As[16 * LD];        // 16.25 KB

    const int tid  = threadIdx.x;
    const int wave = tid >> 5;
    const int lane = tid & 31;
    const int ln   = lane & 15;
    const int hi   = lane >> 4;
    const int e0   = blockIdx.x * 16;    // edge block
    const int bb   = blockIdx.y;         // batch
    const int n0   = wave * 16;          // column block

    {   // cooperative gather: 16 threads per edge row, 2 b128 per side
        const int r  = tid >> 4;         // 0..15 edge within block
        const int c0 = (tid & 15) * 8;   // 0..120
        const int s  = edge_src[e0 + r];
        const int d  = edge_dst[e0 + r];
        const float* srow = img_ws + ((size_t)bb * Nn + s) * Hc;
        const float* drow = img_ws + ((size_t)bb * Nn + d) * Hc;
        copy_b128_to_lds(srow + c0,     &As[r * LD + c0]);
        copy_b128_to_lds(srow + c0 + 4, &As[r * LD + c0 + 4]);
        copy_b128_to_lds(drow + c0,     &As[r * LD + 128 + c0]);
        copy_b128_to_lds(drow + c0 + 4, &As[r * LD + 128 + c0 + 4]);
    }
    wait_async_lds();
    __syncthreads();

    v8f acc = {};
    #pragma unroll 8
    for (int k = 0; k < 2 * Hc; k += 4) {
        v2f a = *(const v2f*)(&As[ln * LD + k + 2 * hi]);
        v2f bf;
        bf.x = W_rel[(size_t)(k     + 2 * hi) * Hc + n0 + ln];
        bf.y = W_rel[(size_t)(k + 1 + 2 * hi) * Hc + n0 + ln];
        acc = __builtin_amdgcn_wmma_f32_16x16x4_f32(
                  false, a, false, bf, (short)0, acc, false, false);
    }

    const int   col = n0 + ln;
    const float br  = b_rel[col];
    #pragma unroll
    for (int j = 0; j < 8; ++j) {
        const int e = e0 + j + 8 * hi;
        edge_out[((size_t)bb * En + e) * Hc + col] = acc[j] + br;
    }
}

// ---------------------------------------------------------------------------
extern "C" void kernel_launch(void* const* d_in, const int* in_sizes, int n_in,
                              void* d_out, int out_size, void* d_ws, size_t ws_size,
                              hipStream_t stream) {
    const float* img_feat = (const float*)d_in[0];
    const float* img_loc  = (const float*)d_in[1];
    const int*   edge_src = (const int*)  d_in[2];
    const int*   edge_dst = (const int*)  d_in[3];
    const float* W_img    = (const float*)d_in[4];
    const float* b_img    = (const float*)d_in[5];
    const float* W_loc    = (const float*)d_in[6];
    const float* b_loc    = (const float*)d_in[7];
    const float* W_rel    = (const float*)d_in[8];
    const float* b_rel    = (const float*)d_in[9];

    float* node_out = (float*)d_out;                          // [B,N,H]
    float* edge_out = (float*)d_out + (size_t)Bn * Nn * Hc;   // [B,E,H]
    float* img_ws   = (float*)d_ws;                           // [B,N,H] scratch (2 MB)

    node_proj_kernel<<<dim3(Bn * Nn / 16), dim3(256), 0, stream>>>(
        img_feat, img_loc, W_img, b_img, W_loc, b_loc, img_ws, node_out);

    edge_kernel<<<dim3(En / 16, Bn), dim3(256), 0, stream>>>(
        img_ws, edge_src, edge_dst, W_rel, b_rel, edge_out);
}